// GCN_90331752169730
// MI455X (gfx1250) — compile-verified
//
#include <hip/hip_runtime.h>
#include <hip/hip_bf16.h>

typedef __attribute__((ext_vector_type(2))) float v2f;
typedef __attribute__((ext_vector_type(8))) float v8f;

#define N_NODES 89250
#define N_EDGES 900000
#define F_IN 500
#define N_CLASSES 7

// ---------------------------------------------------------------------------
// Degree accumulation: out_deg over src, in_deg over dst (float atomics, L2).
// ---------------------------------------------------------------------------
__global__ void gcn_deg_kernel(const int* __restrict__ ei,
                               float* __restrict__ outDeg,
                               float* __restrict__ inDeg, int nEdges) {
    int e = blockIdx.x * blockDim.x + threadIdx.x;
    if (e < nEdges) {
        atomicAdd(&outDeg[ei[e]], 1.0f);
        atomicAdd(&inDeg[ei[nEdges + e]], 1.0f);
    }
}

// norm = rsqrt(max(deg, 1))  (in place)
__global__ void gcn_norm_kernel(float* __restrict__ d, int n) {
    int i = blockIdx.x * blockDim.x + threadIdx.x;
    if (i < n) d[i] = rsqrtf(fmaxf(d[i], 1.0f));
}

// ---------------------------------------------------------------------------
// Pack W [K x Ncols] into per-lane B fragments for V_WMMA_F32_16X16X4_F32.
// Chunk c covers K-rows 4c..4c+3.  Lane l: n = l%16, kOff = (l/16)*2.
//   frag.x = W[4c+kOff][n], frag.y = W[4c+kOff+1][n]   (zero padded)
// ---------------------------------------------------------------------------
__global__ void gcn_pack_b_kernel(const float* __restrict__ W, int K, int Ncols,
                                  v2f* __restrict__ Bpack, int kChunks) {
    int t = blockIdx.x * blockDim.x + threadIdx.x;
    if (t >= kChunks * 32) return;
    int chunk = t >> 5;
    int lane  = t & 31;
    int n = lane & 15;
    int k = chunk * 4 + ((lane >> 4) << 1);
    v2f b;
    b[0] = (k     < K && n < Ncols) ? W[k * Ncols + n]       : 0.0f;
    b[1] = (k + 1 < K && n < Ncols) ? W[(k + 1) * Ncols + n] : 0.0f;
    Bpack[t] = b;
}

// ---------------------------------------------------------------------------
// Fp32 WMMA GEMM:  Xpad[r, 0:16] = (A[r, :] * normSrc[r]) @ Wpack
// One wave per 16-row tile; K consumed 4 at a time via v_wmma_f32_16x16x4_f32.
// A-fragment layout: lane l holds (m = l%16, k = base + (l/16)*2 .. +1).
// D layout: VGPR j -> row (l/16)*8 + j, col l%16.
// ---------------------------------------------------------------------------
__global__ void gcn_wmma_gemm_kernel(const float* __restrict__ A, int lda,
                                     int kChunks, const v2f* __restrict__ Bpack,
                                     const float* __restrict__ normSrc,
                                     float* __restrict__ Xpad, int nRows) {
    int wave = (blockIdx.x * blockDim.x + threadIdx.x) >> 5;
    int lane = threadIdx.x & 31;
    int row0 = wave * 16;
    if (row0 >= nRows) return;                 // wave-uniform exit, EXEC stays full

    int m    = lane & 15;
    int kOff = (lane >> 4) << 1;               // 0 or 2
    int row  = row0 + m;
    int rowC = (row < nRows) ? row : (nRows - 1);   // clamp tail (stores guarded)
    float ns = normSrc[rowC];
    const float* aRow = A + (long)rowC * lda;

    v8f c = {};
    for (int kc = 0; kc < kChunks; ++kc) {
        int k = kc * 4 + kOff;
        v2f a = *(const v2f*)(aRow + k);       // global_load_b64
        a[0] *= ns;
        a[1] *= ns;
        v2f b = Bpack[kc * 32 + lane];         // 256B coalesced, L2-resident
        c = __builtin_amdgcn_wmma_f32_16x16x4_f32(
                false, a, false, b, (short)0, c, false, false);
    }

    int n     = lane & 15;
    int rbase = row0 + ((lane >> 4) << 3);
    for (int j = 0; j < 8; ++j) {
        int r = rbase + j;
        if (r < nRows) Xpad[r * 16 + n] = c[j];
    }
}

// ---------------------------------------------------------------------------
// Edge scatter: agg[dst, c] += x[src, c]; 8 lanes per edge (c = 0..6).
// x (5.7MB) and agg (2.9MB) live in L2, atomics resolve there.
// ---------------------------------------------------------------------------
__global__ void gcn_scatter_kernel(const int* __restrict__ ei,
                                   const float* __restrict__ xpad,
                                   float* __restrict__ agg, int nEdges) {
    int t = blockIdx.x * blockDim.x + threadIdx.x;
    int e = t >> 3;
    int c = t & 7;
    if (e < nEdges && c < N_CLASSES) {
        int src = ei[e];
        int dst = ei[nEdges + e];
        atomicAdd(&agg[dst * 8 + c], xpad[src * 16 + c]);
    }
}

// h1pad[i, c] = agg[i, c] * normDst[i] + bias[c]  (cols 7..15 zeroed for K=8 pad)
__global__ void gcn_finish_pad_kernel(const float* __restrict__ agg,
                                      const float* __restrict__ normDst,
                                      const float* __restrict__ bias,
                                      float* __restrict__ h1pad, int n) {
    int t = blockIdx.x * blockDim.x + threadIdx.x;
    if (t >= n * 16) return;
    int i = t >> 4;
    int c = t & 15;
    float v = 0.0f;
    if (c < N_CLASSES) v = agg[i * 8 + c] * normDst[i] + bias[c];
    h1pad[t] = v;
}

// out[i, c] = agg2[i, c] * normDst[i] + b2[c]   (tight [N,7] output)
__global__ void gcn_final_kernel(const float* __restrict__ agg,
                                 const float* __restrict__ normDst,
                                 const float* __restrict__ bias,
                                 float* __restrict__ out, int n) {
    int t = blockIdx.x * blockDim.x + threadIdx.x;
    int i = t >> 3;
    int c = t & 7;
    if (i < n && c < N_CLASSES)
        out[i * N_CLASSES + c] = agg[i * 8 + c] * normDst[i] + bias[c];
}

static inline size_t alignUp256(size_t x) { return (x + 255) & ~(size_t)255; }

extern "C" void kernel_launch(void* const* d_in, const int* in_sizes, int n_in,
                              void* d_out, int out_size, void* d_ws, size_t ws_size,
                              hipStream_t stream) {
    const float* h  = (const float*)d_in[0];
    const int*   ei = (const int*)  d_in[1];
    const float* W1 = (const float*)d_in[2];
    const float* b1 = (const float*)d_in[3];
    const float* W2 = (const float*)d_in[4];
    const float* b2 = (const float*)d_in[5];
    float* out = (float*)d_out;

    const int N = N_NODES, E = N_EDGES;
    const int kChunks1 = F_IN / 4;   // 125 (exact)
    const int kChunks2 = 2;          // K=7 padded to 8

    // ---- workspace carve (all offsets 256B aligned) ----
    char*  ws  = (char*)d_ws;
    size_t off = 0;
    auto take = [&](size_t bytes) -> void* {
        void* p = ws + off;
        off = alignUp256(off + bytes);
        return p;
    };
    float* normSrc = (float*)take((size_t)N * 4);
    float* normDst = (float*)take((size_t)N * 4);
    float* xpad    = (float*)take((size_t)N * 16 * 4);   // layer1 then layer2 X
    float* h1pad   = (float*)take((size_t)N * 16 * 4);
    float* agg1    = (float*)take((size_t)N * 8 * 4);
    float* agg2    = (float*)take((size_t)N * 8 * 4);
    v2f*   bp1     = (v2f*)take((size_t)kChunks1 * 32 * 8);
    v2f*   bp2     = (v2f*)take((size_t)kChunks2 * 32 * 8);
    (void)ws_size;

    // ---- zero accumulators (capture-legal async memsets) ----
    hipMemsetAsync(normSrc, 0, (size_t)N * 4, stream);
    hipMemsetAsync(normDst, 0, (size_t)N * 4, stream);
    hipMemsetAsync(agg1,    0, (size_t)N * 8 * 4, stream);
    hipMemsetAsync(agg2,    0, (size_t)N * 8 * 4, stream);

    // ---- degrees + norms ----
    gcn_deg_kernel<<<(E + 255) / 256, 256, 0, stream>>>(ei, normSrc, normDst, E);
    gcn_norm_kernel<<<(N + 255) / 256, 256, 0, stream>>>(normSrc, N);
    gcn_norm_kernel<<<(N + 255) / 256, 256, 0, stream>>>(normDst, N);

    // ---- pack weights into WMMA B fragments ----
    gcn_pack_b_kernel<<<(kChunks1 * 32 + 255) / 256, 256, 0, stream>>>(
        W1, F_IN, N_CLASSES, bp1, kChunks1);
    gcn_pack_b_kernel<<<1, 64, 0, stream>>>(
        W2, N_CLASSES, N_CLASSES, bp2, kChunks2);

    // ---- layer 1 ----
    int tiles  = (N + 15) / 16;
    int gemmBl = (tiles * 32 + 255) / 256;
    gcn_wmma_gemm_kernel<<<gemmBl, 256, 0, stream>>>(
        h, F_IN, kChunks1, bp1, normSrc, xpad, N);
    gcn_scatter_kernel<<<(E * 8 + 255) / 256, 256, 0, stream>>>(ei, xpad, agg1, E);
    gcn_finish_pad_kernel<<<(N * 16 + 255) / 256, 256, 0, stream>>>(
        agg1, normDst, b1, h1pad, N);

    // ---- layer 2 ----
    gcn_wmma_gemm_kernel<<<gemmBl, 256, 0, stream>>>(
        h1pad, 16, kChunks2, bp2, normSrc, xpad, N);
    gcn_scatter_kernel<<<(E * 8 + 255) / 256, 256, 0, stream>>>(ei, xpad, agg2, E);
    gcn_final_kernel<<<(N * 8 + 255) / 256, 256, 0, stream>>>(
        agg2, normDst, b2, out, N);

    (void)in_sizes; (void)n_in; (void)out_size;
}